// Graph_Critic_Model_54305566491363
// MI455X (gfx1250) — compile-verified
//
#include <hip/hip_runtime.h>
#include <hip/hip_bf16.h>

// ---------------------------------------------------------------------------
// Graph critic model for MI455X (gfx1250, wave32).
// N=50000 nodes (3125 tiles of 16), H=32, E=1.6M edges.
// WMMA: V_WMMA_F32_16X16X4_F32, one 16-node tile per wave, 2x v8f accumulators.
// Dominant HBM cost: 409.6 MB W_v1 stream -> b128 non-temporal loads.
// ---------------------------------------------------------------------------

#define NN 50000
#define FDIM 128
#define ADIM 16
#define K1 144          // FDIM + ADIM
#define HH 32
#define EE 1600000
#define NTILES 3125     // NN / 16
#define XIN_STRIDE 148  // padded LDS row stride (bank-conflict-free A reads)
#define S_STRIDE 36     // padded LDS stride for 16x32 tiles

typedef __attribute__((ext_vector_type(2))) float v2f;
typedef __attribute__((ext_vector_type(4))) float v4f;
typedef __attribute__((ext_vector_type(8))) float v8f;

static __device__ inline v8f wmma_f32_4(v2f a, v2f b, v8f c) {
  // D = A(16x4) x B(4x16) + C(16x16), fp32 throughout.
  return __builtin_amdgcn_wmma_f32_16x16x4_f32(
      /*neg_a=*/false, a, /*neg_b=*/false, b,
      /*c_mod=*/(short)0, c, /*reuse_a=*/false, /*reuse_b=*/false);
}

static __device__ inline v8f vzero8() {
  v8f z = {0.f, 0.f, 0.f, 0.f, 0.f, 0.f, 0.f, 0.f};
  return z;
}

// ---------------------------------------------------------------------------
// Encoder: X1 = relu(Xin@We1+b1); X2 = relu(X1@We2+b2); Y = X2@Wg
// One wave (32 threads) per 16-node tile.
// ---------------------------------------------------------------------------
__global__ void __launch_bounds__(32)
encoder_kernel(const float* __restrict__ obs, const float* __restrict__ act,
               const float* __restrict__ We1, const float* __restrict__ be1,
               const float* __restrict__ We2, const float* __restrict__ be2,
               const float* __restrict__ Wg,
               float* __restrict__ X, float* __restrict__ Y) {
  __shared__ float xin[16 * XIN_STRIDE];
  __shared__ float x1[16 * S_STRIDE];
  __shared__ float x2[16 * S_STRIDE];

  const int lane = threadIdx.x;
  const int base = blockIdx.x * 16;
  const int ml = lane & 15;     // M row (also N col for B frags)
  const int hi = lane >> 4;     // upper half-wave -> K+2

  // Stage obs tile: 16 contiguous rows x 128 floats = 512 float4, coalesced.
  {
    const float4* o4 = (const float4*)(obs + (size_t)base * FDIM);
    for (int i = lane; i < 512; i += 32) {
      int r = i >> 5, c4 = i & 31;
      *(float4*)&xin[r * XIN_STRIDE + c4 * 4] = o4[i];
    }
    const float4* a4 = (const float4*)(act + (size_t)base * ADIM);
    for (int i = lane; i < 64; i += 32) {
      int r = i >> 2, c4 = i & 3;
      *(float4*)&xin[r * XIN_STRIDE + FDIM + c4 * 4] = a4[i];
    }
  }
  __syncthreads();

  // ---- GEMM1: [16 x 144] @ [144 x 32] ----
  v8f c0 = vzero8(), c1 = vzero8();
  for (int k0 = 0; k0 < K1; k0 += 4) {
    int k = k0 + 2 * hi;
    v2f a;  a.x = xin[ml * XIN_STRIDE + k];     a.y = xin[ml * XIN_STRIDE + k + 1];
    v2f b0; b0.x = We1[k * HH + ml];            b0.y = We1[(k + 1) * HH + ml];
    v2f b1; b1.x = We1[k * HH + 16 + ml];       b1.y = We1[(k + 1) * HH + 16 + ml];
    c0 = wmma_f32_4(a, b0, c0);
    c1 = wmma_f32_4(a, b1, c1);
  }
  {
    float bn0 = be1[ml], bn1 = be1[ml + 16];
    for (int r = 0; r < 8; ++r) {
      int row = r + 8 * hi;
      x1[row * S_STRIDE + ml]      = fmaxf(c0[r] + bn0, 0.f);
      x1[row * S_STRIDE + 16 + ml] = fmaxf(c1[r] + bn1, 0.f);
    }
  }
  __syncthreads();

  // ---- GEMM2: [16 x 32] @ [32 x 32] ----
  c0 = vzero8(); c1 = vzero8();
  for (int k0 = 0; k0 < HH; k0 += 4) {
    int k = k0 + 2 * hi;
    v2f a;  a.x = x1[ml * S_STRIDE + k];        a.y = x1[ml * S_STRIDE + k + 1];
    v2f b0; b0.x = We2[k * HH + ml];            b0.y = We2[(k + 1) * HH + ml];
    v2f b1; b1.x = We2[k * HH + 16 + ml];       b1.y = We2[(k + 1) * HH + 16 + ml];
    c0 = wmma_f32_4(a, b0, c0);
    c1 = wmma_f32_4(a, b1, c1);
  }
  {
    float bn0 = be2[ml], bn1 = be2[ml + 16];
    for (int r = 0; r < 8; ++r) {
      int row = r + 8 * hi;
      float v0 = fmaxf(c0[r] + bn0, 0.f);
      float v1 = fmaxf(c1[r] + bn1, 0.f);
      x2[row * S_STRIDE + ml] = v0;
      x2[row * S_STRIDE + 16 + ml] = v1;
      X[(size_t)(base + row) * HH + ml] = v0;
      X[(size_t)(base + row) * HH + 16 + ml] = v1;
    }
  }
  __syncthreads();

  // ---- GEMM3: Y = X2 @ Wg (no bias/relu here; applied after aggregation) ----
  c0 = vzero8(); c1 = vzero8();
  for (int k0 = 0; k0 < HH; k0 += 4) {
    int k = k0 + 2 * hi;
    v2f a;  a.x = x2[ml * S_STRIDE + k];        a.y = x2[ml * S_STRIDE + k + 1];
    v2f b0; b0.x = Wg[k * HH + ml];             b0.y = Wg[(k + 1) * HH + ml];
    v2f b1; b1.x = Wg[k * HH + 16 + ml];        b1.y = Wg[(k + 1) * HH + 16 + ml];
    c0 = wmma_f32_4(a, b0, c0);
    c1 = wmma_f32_4(a, b1, c1);
  }
  for (int r = 0; r < 8; ++r) {
    int row = r + 8 * hi;
    Y[(size_t)(base + row) * HH + ml] = c0[r];
    Y[(size_t)(base + row) * HH + 16 + ml] = c1[r];
  }
}

// ---------------------------------------------------------------------------
// Degree / normalization / aggregation
// ---------------------------------------------------------------------------
__global__ void deg_init_kernel(float* __restrict__ deg) {
  int i = blockIdx.x * blockDim.x + threadIdx.x;
  if (i < NN) deg[i] = 1.0f;  // self-loop folded in
}

__global__ void edge_deg_kernel(const int* __restrict__ ei, float* __restrict__ deg) {
  int e = blockIdx.x * blockDim.x + threadIdx.x;
  if (e < EE) atomicAdd(&deg[ei[EE + e]], 1.0f);  // col = target
}

// dinv + self-loop contribution seeds agg (writes every element; no pre-zero).
__global__ void node_init_kernel(const float* __restrict__ deg,
                                 const float* __restrict__ Y,
                                 float* __restrict__ dinv,
                                 float* __restrict__ agg) {
  int t = blockIdx.x * blockDim.x + threadIdx.x;
  if (t >= NN * HH) return;
  int i = t >> 5, f = t & 31;
  float d = rsqrtf(deg[i]);   // deg >= 1 always
  if (f == 0) dinv[i] = d;
  agg[t] = Y[t] * d * d;
}

// 8 threads per edge, one float4 each; atomics land in L2-resident agg.
__global__ void edge_scatter_kernel(const int* __restrict__ ei,
                                    const float* __restrict__ dinv,
                                    const float* __restrict__ Y,
                                    float* __restrict__ agg) {
  int t = blockIdx.x * blockDim.x + threadIdx.x;
  int e = t >> 3, q = t & 7;
  if (e >= EE) return;
  int r = ei[e];         // source
  int c = ei[EE + e];    // target
  float nrm = dinv[r] * dinv[c];
  float4 y = *(const float4*)(Y + (size_t)r * HH + q * 4);
  float* dst = agg + (size_t)c * HH + q * 4;
  atomicAdd(dst + 0, y.x * nrm);
  atomicAdd(dst + 1, y.y * nrm);
  atomicAdd(dst + 2, y.z * nrm);
  atomicAdd(dst + 3, y.w * nrm);
}

// ---------------------------------------------------------------------------
// Post: Xg=relu(agg+bg); Xg2=relu(Xg@Wgd+bgd); Xp1=relu([Xg2|X]@Wp1+bp1);
//       Xp2=relu(Xp1@Wp2+bp2). One wave per 16-node tile.
// ---------------------------------------------------------------------------
__global__ void __launch_bounds__(32)
post_kernel(const float* __restrict__ agg, const float* __restrict__ X,
            const float* __restrict__ bg,
            const float* __restrict__ Wgd, const float* __restrict__ bgd,
            const float* __restrict__ Wp1, const float* __restrict__ bp1,
            const float* __restrict__ Wp2, const float* __restrict__ bp2,
            float* __restrict__ Xp) {
  __shared__ float s1[16 * S_STRIDE];
  __shared__ float s2[16 * S_STRIDE];

  const int lane = threadIdx.x;
  const int base = blockIdx.x * 16;
  const int ml = lane & 15;
  const int hi = lane >> 4;

  // ---- Xg2 = relu( relu(agg+bg) @ Wgd + bgd ) ----
  v8f c0 = vzero8(), c1 = vzero8();
  for (int k0 = 0; k0 < HH; k0 += 4) {
    int k = k0 + 2 * hi;
    v2f a;
    a.x = fmaxf(agg[(size_t)(base + ml) * HH + k] + bg[k], 0.f);
    a.y = fmaxf(agg[(size_t)(base + ml) * HH + k + 1] + bg[k + 1], 0.f);
    v2f b0; b0.x = Wgd[k * HH + ml];       b0.y = Wgd[(k + 1) * HH + ml];
    v2f b1; b1.x = Wgd[k * HH + 16 + ml];  b1.y = Wgd[(k + 1) * HH + 16 + ml];
    c0 = wmma_f32_4(a, b0, c0);
    c1 = wmma_f32_4(a, b1, c1);
  }
  {
    float bn0 = bgd[ml], bn1 = bgd[ml + 16];
    for (int r = 0; r < 8; ++r) {
      int row = r + 8 * hi;
      s1[row * S_STRIDE + ml]      = fmaxf(c0[r] + bn0, 0.f);
      s1[row * S_STRIDE + 16 + ml] = fmaxf(c1[r] + bn1, 0.f);
    }
  }
  __syncthreads();

  // ---- Xp1 = relu( Xg2 @ Wp1[0:32] + X @ Wp1[32:64] + bp1 ) ----
  c0 = vzero8(); c1 = vzero8();
  for (int k0 = 0; k0 < HH; k0 += 4) {
    int k = k0 + 2 * hi;
    v2f a;  a.x = s1[ml * S_STRIDE + k];   a.y = s1[ml * S_STRIDE + k + 1];
    v2f b0; b0.x = Wp1[k * HH + ml];       b0.y = Wp1[(k + 1) * HH + ml];
    v2f b1; b1.x = Wp1[k * HH + 16 + ml];  b1.y = Wp1[(k + 1) * HH + 16 + ml];
    c0 = wmma_f32_4(a, b0, c0);
    c1 = wmma_f32_4(a, b1, c1);
  }
  for (int k0 = 0; k0 < HH; k0 += 4) {
    int k = k0 + 2 * hi;
    v2f a;
    a.x = X[(size_t)(base + ml) * HH + k];
    a.y = X[(size_t)(base + ml) * HH + k + 1];
    int kk = 32 + k;
    v2f b0; b0.x = Wp1[kk * HH + ml];       b0.y = Wp1[(kk + 1) * HH + ml];
    v2f b1; b1.x = Wp1[kk * HH + 16 + ml];  b1.y = Wp1[(kk + 1) * HH + 16 + ml];
    c0 = wmma_f32_4(a, b0, c0);
    c1 = wmma_f32_4(a, b1, c1);
  }
  {
    float bn0 = bp1[ml], bn1 = bp1[ml + 16];
    for (int r = 0; r < 8; ++r) {
      int row = r + 8 * hi;
      s2[row * S_STRIDE + ml]      = fmaxf(c0[r] + bn0, 0.f);
      s2[row * S_STRIDE + 16 + ml] = fmaxf(c1[r] + bn1, 0.f);
    }
  }
  __syncthreads();

  // ---- Xp2 = relu( Xp1 @ Wp2 + bp2 ) ----
  c0 = vzero8(); c1 = vzero8();
  for (int k0 = 0; k0 < HH; k0 += 4) {
    int k = k0 + 2 * hi;
    v2f a;  a.x = s2[ml * S_STRIDE + k];   a.y = s2[ml * S_STRIDE + k + 1];
    v2f b0; b0.x = Wp2[k * HH + ml];       b0.y = Wp2[(k + 1) * HH + ml];
    v2f b1; b1.x = Wp2[k * HH + 16 + ml];  b1.y = Wp2[(k + 1) * HH + 16 + ml];
    c0 = wmma_f32_4(a, b0, c0);
    c1 = wmma_f32_4(a, b1, c1);
  }
  {
    float bn0 = bp2[ml], bn1 = bp2[ml + 16];
    for (int r = 0; r < 8; ++r) {
      int row = r + 8 * hi;
      Xp[(size_t)(base + row) * HH + ml]      = fmaxf(c0[r] + bn0, 0.f);
      Xp[(size_t)(base + row) * HH + 16 + ml] = fmaxf(c1[r] + bn1, 0.f);
    }
  }
}

// ---------------------------------------------------------------------------
// Final matvec: V64 = W_v1^T @ xp  (409.6 MB stream -> HBM-bound).
// b128 non-temporal loads: lane owns a 16B column slice, 16 lanes = one
// 256B row of W_v1. 16 row-groups per block, LDS linear reduce, 64 atomics.
// ---------------------------------------------------------------------------
__global__ void v64_init_kernel(float* __restrict__ V64) {
  if (threadIdx.x < 64) V64[threadIdx.x] = 0.f;
}

__global__ void __launch_bounds__(256)
matvec_kernel(const float* __restrict__ Xp, const float* __restrict__ Wv1,
              float* __restrict__ V64) {
  const int jg = threadIdx.x & 15;   // column group: floats 4*jg .. 4*jg+3
  const int g  = threadIdx.x >> 4;   // row group 0..15
  const int rows = NN * HH;          // 1.6M
  v4f acc = {0.f, 0.f, 0.f, 0.f};
  for (int row = blockIdx.x * 16 + g; row < rows; row += gridDim.x * 16) {
    const v4f* wp = (const v4f*)(Wv1 + (size_t)row * 64) + jg;
    v4f w = __builtin_nontemporal_load(wp);   // streamed once, keep L2 clean
    acc += w * Xp[row];
  }
  __shared__ v4f red[256];
  red[threadIdx.x] = acc;
  __syncthreads();
  if (threadIdx.x < 16) {
    v4f s = red[threadIdx.x];
    for (int gg = 1; gg < 16; ++gg) s += red[gg * 16 + threadIdx.x];
    float* dst = V64 + threadIdx.x * 4;
    atomicAdd(dst + 0, s[0]);
    atomicAdd(dst + 1, s[1]);
    atomicAdd(dst + 2, s[2]);
    atomicAdd(dst + 3, s[3]);
  }
}

__global__ void final_kernel(const float* __restrict__ V64,
                             const float* __restrict__ bv1,
                             const float* __restrict__ Wv2,
                             const float* __restrict__ bv2,
                             float* __restrict__ out) {
  __shared__ float s[64];
  int t = threadIdx.x;
  float v = fmaxf(V64[t] + bv1[t], 0.f);
  s[t] = v * Wv2[t];
  __syncthreads();
  if (t == 0) {
    float acc = bv2[0];
    for (int i = 0; i < 64; ++i) acc += s[i];
    out[0] = acc;
  }
}

// ---------------------------------------------------------------------------
extern "C" void kernel_launch(void* const* d_in, const int* in_sizes, int n_in,
                              void* d_out, int out_size, void* d_ws, size_t ws_size,
                              hipStream_t stream) {
  const float* obs = (const float*)d_in[0];
  const float* act = (const float*)d_in[1];
  const int*   ei  = (const int*)d_in[2];   // [2, E] int32 (jax default x64-off)
  const float* We1 = (const float*)d_in[3];
  const float* be1 = (const float*)d_in[4];
  const float* We2 = (const float*)d_in[5];
  const float* be2 = (const float*)d_in[6];
  const float* Wg  = (const float*)d_in[7];
  const float* bg  = (const float*)d_in[8];
  const float* Wgd = (const float*)d_in[9];
  const float* bgd = (const float*)d_in[10];
  const float* Wp1 = (const float*)d_in[11];
  const float* bp1 = (const float*)d_in[12];
  const float* Wp2 = (const float*)d_in[13];
  const float* bp2 = (const float*)d_in[14];
  const float* Wv1 = (const float*)d_in[15];
  const float* bv1 = (const float*)d_in[16];
  const float* Wv2 = (const float*)d_in[17];
  const float* bv2 = (const float*)d_in[18];
  float* out = (float*)d_out;

  // Workspace layout (floats): X, Y, agg, Xp [N*H each], deg, dinv [N], V64.
  float* ws   = (float*)d_ws;
  float* X    = ws;                       // 1,600,000
  float* Y    = X + (size_t)NN * HH;      // 1,600,000
  float* agg  = Y + (size_t)NN * HH;      // 1,600,000
  float* Xp   = agg + (size_t)NN * HH;    // 1,600,000
  float* deg  = Xp + (size_t)NN * HH;     // 50,000
  float* dinv = deg + NN;                 // 50,000
  float* V64  = dinv + NN;                // 64

  // 1. Encoder (X, Y)
  encoder_kernel<<<NTILES, 32, 0, stream>>>(obs, act, We1, be1, We2, be2, Wg, X, Y);

  // 2. Degrees + normalization + aggregation
  deg_init_kernel<<<(NN + 255) / 256, 256, 0, stream>>>(deg);
  edge_deg_kernel<<<(EE + 255) / 256, 256, 0, stream>>>(ei, deg);
  node_init_kernel<<<(NN * HH + 255) / 256, 256, 0, stream>>>(deg, Y, dinv, agg);
  edge_scatter_kernel<<<(EE * 8 + 255) / 256, 256, 0, stream>>>(ei, dinv, Y, agg);

  // 3. Post-GCN MLP chain (Xp)
  post_kernel<<<NTILES, 32, 0, stream>>>(agg, X, bg, Wgd, bgd, Wp1, bp1, Wp2, bp2, Xp);

  // 4. Big matvec (bandwidth-bound) + head
  v64_init_kernel<<<1, 64, 0, stream>>>(V64);
  matvec_kernel<<<2048, 256, 0, stream>>>(Xp, Wv1, V64);
  final_kernel<<<1, 64, 0, stream>>>(V64, bv1, Wv2, bv2, out);
}